// CTCLayer_72129680769726
// MI455X (gfx1250) — compile-verified
//
#include <hip/hip_runtime.h>
#include <math.h>
#include <stdint.h>

// ---------------- problem constants (from reference setup_inputs) ------------
#define CTC_T 512
#define CTC_C 128
#define CTC_L 32
#define CTC_S (2 * CTC_L + 1)          // 65 extended states
#define CHUNK 8                        // y_pred rows per async prefetch chunk
#define NCHUNK (CTC_T / CHUNK)
#define NEG_INF (-1e30f)
#define CTC_EPS (1e-7f)

static_assert(CTC_T % CHUNK == 0, "chunking");

// ---------------- CDNA5 async global->LDS path (guarded) --------------------
#if __has_builtin(__builtin_amdgcn_global_load_async_to_lds_b128)
#define HAS_ASYNC_LDS 1
#else
#define HAS_ASYNC_LDS 0
#endif

typedef int v4i __attribute__((vector_size(16)));
typedef __attribute__((address_space(1))) v4i gv4i_t;   // global int4
typedef __attribute__((address_space(3))) v4i lv4i_t;   // LDS int4

#define CFENCE() asm volatile("" ::: "memory")

#if HAS_ASYNC_LDS
#if __has_builtin(__builtin_amdgcn_s_wait_asynccnt)
#define WAIT_ASYNC(n) do { __builtin_amdgcn_s_wait_asynccnt(n); CFENCE(); } while (0)
#else
#define WAIT_ASYNC(n) asm volatile("s_wait_asynccnt %0" :: "n"(n) : "memory")
#endif
#else
#define WAIT_ASYNC(n) ((void)0)
#endif

#if __has_builtin(__builtin_amdgcn_s_wait_dscnt)
#define WAIT_DS0() do { __builtin_amdgcn_s_wait_dscnt(0); CFENCE(); } while (0)
#else
#define WAIT_DS0() asm volatile("s_wait_dscnt 0" ::: "memory")
#endif

__device__ __forceinline__ void async_row_load(const float* gsrc, float* lds_dst) {
#if HAS_ASYNC_LDS
    // one lane moves 16 B; 32 lanes move one 512 B row of y_pred
    __builtin_amdgcn_global_load_async_to_lds_b128(
        (gv4i_t*)gsrc, (lv4i_t*)lds_dst, 0, 0);
#else
    (void)gsrc; (void)lds_dst;
#endif
}

// ---------------- kernel: one wave32 per batch element ----------------------
__global__ __launch_bounds__(32)
void ctc_fwd_kernel(const int* __restrict__ y_true,
                    const float* __restrict__ y_pred,
                    float* __restrict__ out) {
    const int b    = blockIdx.x;
    const int lane = threadIdx.x;

    __shared__ __align__(16) float rows[2][CHUNK][CTC_C];   // 8 KB double buffer
    __shared__ float alpha[CTC_S + 2];                      // [0..1] = -inf pad

    // Per-lane state metadata (registers): lane handles s = lane + 32k.
    const int* lab = y_true + b * CTC_L;
    int  cls[3];
    bool skp[3];
    bool act[3];
#pragma unroll
    for (int k = 0; k < 3; ++k) {
        const int s = lane + 32 * k;
        act[k] = (s < CTC_S);
        int  c  = CTC_C - 1;    // blank
        bool sk = false;
        if (act[k] && (s & 1)) {
            const int i = s >> 1;
            c  = lab[i];
            sk = (i > 0) && (c != lab[i - 1]);
        }
        cls[k] = c;
        skp[k] = sk;
    }
    if (lane < 2) alpha[lane] = NEG_INF;   // pad for s-1 / s-2 reads
    CFENCE();

    const float* gbase = y_pred + (size_t)b * CTC_T * CTC_C;

#if HAS_ASYNC_LDS
    // Prologue: chunks 0 and 1 in flight (ASYNCcnt = 16).
#pragma unroll
    for (int r = 0; r < CHUNK; ++r)
        async_row_load(gbase + r * CTC_C + lane * 4, &rows[0][r][lane * 4]);
#pragma unroll
    for (int r = 0; r < CHUNK; ++r)
        async_row_load(gbase + (CHUNK + r) * CTC_C + lane * 4, &rows[1][r][lane * 4]);
#endif

    for (int ci = 0; ci < NCHUNK; ++ci) {
        const int buf = ci & 1;

#if HAS_ASYNC_LDS
        // Async loads complete in order: <=CHUNK outstanding => chunk ci landed.
        if (ci + 1 < NCHUNK) { WAIT_ASYNC(CHUNK); } else { WAIT_ASYNC(0); }
#else
        // Fallback: synchronous staged copy + prefetch of next chunk into caches.
        const float* gc = gbase + (size_t)ci * CHUNK * CTC_C;
#pragma unroll
        for (int r = 0; r < CHUNK; ++r) {
            const float4 v = *(const float4*)(gc + r * CTC_C + lane * 4);
            *(float4*)(&rows[buf][r][lane * 4]) = v;
        }
        if (ci + 1 < NCHUNK) {
            const float* gn = gbase + (size_t)(ci + 1) * CHUNK * CTC_C;
#pragma unroll
            for (int r = 0; r < CHUNK; ++r)
                __builtin_prefetch(gn + r * CTC_C + lane * 4, 0, 1);
        }
        CFENCE();
#endif

        // ---- Phase A: gather + log for the whole chunk (independent of the
        //      alpha recurrence). All rows[buf] LDS reads retire here, fully
        //      pipelined; results live in 24 VGPRs.
        float lp[CHUNK][3];
#pragma unroll
        for (int rr = 0; rr < CHUNK; ++rr) {
            const float* row = rows[buf][rr];
#pragma unroll
            for (int k = 0; k < 3; ++k)
                lp[rr][k] = act[k] ? __logf(row[cls[k]] + CTC_EPS) : 0.0f;
        }
        CFENCE();

        // ---- Phase B: 8 strictly sequential alpha updates; only alpha LDS
        //      traffic remains on the dependency chain.
#pragma unroll
        for (int rr = 0; rr < CHUNK; ++rr) {
            const int t = ci * CHUNK + rr;
            float na[3];
            // read previous alpha (all lanes, all k) and compute
#pragma unroll
            for (int k = 0; k < 3; ++k) {
                na[k] = NEG_INF;
                const int s = lane + 32 * k;
                if (act[k]) {
                    if (t == 0) {
                        na[k] = (s < 2) ? lp[rr][k] : NEG_INF;
                    } else {
                        const float a0 = alpha[2 + s];
                        const float a1 = alpha[1 + s];
                        const float a2 = skp[k] ? alpha[s] : NEG_INF;
                        const float m  = fmaxf(a0, fmaxf(a1, a2));
                        const float sum = __expf(a0 - m) + __expf(a1 - m) + __expf(a2 - m);
                        na[k] = m + __logf(sum) + lp[rr][k];
                    }
                }
            }
            CFENCE();   // keep every cross-lane read before any write (in-order LDS pipe)
#pragma unroll
            for (int k = 0; k < 3; ++k) {
                const int s = lane + 32 * k;
                if (act[k]) alpha[2 + s] = na[k];
            }
            CFENCE();
        }

#if HAS_ASYNC_LDS
        if (ci + 2 < NCHUNK) {
            WAIT_DS0();   // LDS reads of rows[buf] retired before async engine rewrites it
            const float* gn = gbase + (size_t)(ci + 2) * CHUNK * CTC_C;
#pragma unroll
            for (int r = 0; r < CHUNK; ++r)
                async_row_load(gn + r * CTC_C + lane * 4, &rows[buf][r][lane * 4]);
        }
#endif
    }

    if (lane == 0) {
        const float a1 = alpha[2 + CTC_S - 1];
        const float a2 = alpha[2 + CTC_S - 2];
        const float m  = fmaxf(a1, a2);
        const float ll = m + __logf(__expf(a1 - m) + __expf(a2 - m));
        out[b] = -ll;   // keras ctc_batch_cost: negative log-likelihood, [B,1]
    }
}

// ---------------- host-side launch ------------------------------------------
extern "C" void kernel_launch(void* const* d_in, const int* in_sizes, int n_in,
                              void* d_out, int out_size, void* d_ws, size_t ws_size,
                              hipStream_t stream) {
    (void)in_sizes; (void)n_in; (void)d_ws; (void)ws_size;
    const int*   y_true = (const int*)d_in[0];     // [B, L] int32
    const float* y_pred = (const float*)d_in[1];   // [B, T, C] float32
    float*       out    = (float*)d_out;           // [B, 1] float32
    const int B = out_size;                        // 512
    ctc_fwd_kernel<<<B, 32, 0, stream>>>(y_true, y_pred, out);
}